// AutoTransformer_27230092656858
// MI455X (gfx1250) — compile-verified
//
#include <hip/hip_runtime.h>

typedef __attribute__((ext_vector_type(2))) float v2f;
typedef __attribute__((ext_vector_type(8))) float v8f;

#define S_DIM   512
#define B_DIM   32
#define C_DIM   1024
#define K_HEADS 4
#define N_TOK   (S_DIM * B_DIM)   // 16384
#define A_TOT   5632
#define LSTR    17                // 16 + 1 pad -> conflict-free LDS columns
#define HALF_SZ (C_DIM * LSTR)    // floats per 16-row half-tile buffer
#define M_TILE  32

// ---------------------------------------------------------------------------
// Kernel 1: reset workspace counters (ws is not re-initialized between calls).
// ---------------------------------------------------------------------------
__global__ void init_kernel(int* __restrict__ count, int* __restrict__ minT) {
  const int i = threadIdx.x;
  if (i < K_HEADS) count[i] = 0;
  if (i < B_DIM)   minT[i]  = 0x7fffffff;
}

// ---------------------------------------------------------------------------
// Kernel 2: bucket tokens by readout type (compaction) + per-column min_t.
// ---------------------------------------------------------------------------
__global__ void classify_kernel(const int* __restrict__ rx,
                                const int* __restrict__ rt,
                                int* __restrict__ count,
                                int* __restrict__ minT,
                                int* __restrict__ list) {
  const int t = blockIdx.x * 256 + threadIdx.x;
  if (t >= N_TOK) return;
  const int k = rx[t];
  const int pos = atomicAdd(&count[k], 1);
  list[k * N_TOK + pos] = t;
  atomicMin(&minT[t & (B_DIM - 1)], rt[t]);
}

// ---------------------------------------------------------------------------
// Kernel 3: fused  out = scatter( relu(X W1[k] + b1[k]) W2[k] + b2[k] )
// One block = 32 tokens of one head. 256 threads = 8 waves (wave32).
// Each B fragment feeds two M-fragments (rows 0-15 / 16-31) -> half the
// weight (L2) traffic per FLOP vs an M=16 tile.
// ---------------------------------------------------------------------------
__global__ __launch_bounds__(256) void fused_mlp_kernel(
    const float* __restrict__ x,
    const int*   __restrict__ rt,
    const float* __restrict__ W1,
    const float* __restrict__ b1,
    const float* __restrict__ W2_0, const float* __restrict__ b2_0,
    const float* __restrict__ W2_1, const float* __restrict__ b2_1,
    const float* __restrict__ W2_2, const float* __restrict__ b2_2,
    const float* __restrict__ W2_3, const float* __restrict__ b2_3,
    const int* __restrict__ count,
    const int* __restrict__ minT,
    const int* __restrict__ list,
    float* __restrict__ out) {
  const int k   = blockIdx.y;
  const int cnt = count[k];
  const int m0  = blockIdx.x * M_TILE;
  if (m0 >= cnt) return;                       // surplus tile for this head
  const int rows = (cnt - m0 < M_TILE) ? (cnt - m0) : M_TILE;

  const float* W2; const float* b2; int Dk; int off;
  if (k == 0)      { W2 = W2_0; b2 = b2_0; Dk = 2048; off = 0;    }
  else if (k == 1) { W2 = W2_1; b2 = b2_1; Dk = 2048; off = 2048; }
  else if (k == 2) { W2 = W2_2; b2 = b2_2; Dk = 1024; off = 4096; }
  else             { W2 = W2_3; b2 = b2_3; Dk = 512;  off = 5120; }

  // Two stride-17 half-tiles per matrix: [half][c][m&15]; conflict-free A reads.
  __shared__ float Xt[2 * HALF_SZ];
  __shared__ float Ht[2 * HALF_SZ];
  __shared__ int       tok[M_TILE];
  __shared__ long long obase[M_TILE];          // flat output row base

  const int tid = threadIdx.x;
  if (tid < M_TILE) {
    int t = (tid < rows) ? list[k * N_TOK + m0 + tid] : -1;
    tok[tid] = t;
    long long base = 0;
    if (t >= 0) {
      const int b = t & (B_DIM - 1);
      const int i = rt[t] - minT[b];           // reference's readout_i
      base = ((long long)i * B_DIM + b) * A_TOT;
    }
    obase[tid] = base;
  }
  __syncthreads();

  // Stage X rows into LDS transposed; coalesced global reads.
  for (int i = tid; i < M_TILE * C_DIM; i += 256) {
    const int c = i & (C_DIM - 1);
    const int r = i >> 10;
    const int t = tok[r];
    Xt[(r >> 4) * HALF_SZ + c * LSTR + (r & 15)] =
        (t >= 0) ? x[(size_t)t * C_DIM + c] : 0.0f;
  }

  // Zero the non-selected-head slices of the rows this tile owns, so each
  // output row is fully written exactly once (no global memset pass).
  for (int r = 0; r < rows; ++r) {
    const long long base = obase[r];
    for (int a = tid;            a < off;   a += 256) out[base + a] = 0.0f;
    for (int a = off + Dk + tid; a < A_TOT; a += 256) out[base + a] = 0.0f;
  }
  __syncthreads();

  const int lane    = tid & 31;
  const int wave    = tid >> 5;           // 0..7
  const int col     = lane & 15;          // N index (B/D); M index for A reads
  const int kHalf   = (lane >> 4) << 1;   // 0 or 2 : per-lane K sub-offset
  const int rowBase = (lane >> 4) << 3;   // D-fragment row base (0 or 8)

  // ---------------- GEMM1: H = relu(X @ W1[k] + b1[k]) -> Ht ----------------
  const float* W1k = W1 + (size_t)k * C_DIM * C_DIM;
  for (int nt = wave; nt < C_DIM / 16; nt += 8) {
    const int n0 = nt * 16;
    v8f acc0 = {0.f, 0.f, 0.f, 0.f, 0.f, 0.f, 0.f, 0.f};
    v8f acc1 = {0.f, 0.f, 0.f, 0.f, 0.f, 0.f, 0.f, 0.f};
    const float* Bp = W1k + n0 + col;
#pragma unroll 4
    for (int kk = 0; kk < C_DIM / 4; ++kk) {
      const int kb = kk * 4 + kHalf;
      v2f b, a0, a1;
      b.x  = Bp[(size_t)kb * C_DIM];
      b.y  = Bp[(size_t)(kb + 1) * C_DIM];
      a0.x = Xt[kb * LSTR + col];
      a0.y = Xt[(kb + 1) * LSTR + col];
      a1.x = Xt[HALF_SZ + kb * LSTR + col];
      a1.y = Xt[HALF_SZ + (kb + 1) * LSTR + col];
      acc0 = __builtin_amdgcn_wmma_f32_16x16x4_f32(
          false, a0, false, b, (short)0, acc0, false, false);
      acc1 = __builtin_amdgcn_wmma_f32_16x16x4_f32(
          false, a1, false, b, (short)0, acc1, false, false);
    }
    const float bias = b1[k * C_DIM + n0 + col];
    const int hbase = (n0 + col) * LSTR + rowBase;
#pragma unroll
    for (int v = 0; v < 8; ++v) {
      float h0 = fmaxf(acc0[v] + bias, 0.0f);
      float h1 = fmaxf(acc1[v] + bias, 0.0f);
      Ht[hbase + v]           = h0;
      Ht[HALF_SZ + hbase + v] = h1;
    }
  }
  __syncthreads();

  // ---------------- GEMM2: logits = H @ W2[k] + b2[k] -> out slice ----------
  for (int nt = wave; nt < Dk / 16; nt += 8) {
    const int n0 = nt * 16;
    v8f acc0 = {0.f, 0.f, 0.f, 0.f, 0.f, 0.f, 0.f, 0.f};
    v8f acc1 = {0.f, 0.f, 0.f, 0.f, 0.f, 0.f, 0.f, 0.f};
    const float* Bp = W2 + n0 + col;
#pragma unroll 4
    for (int kk = 0; kk < C_DIM / 4; ++kk) {
      const int kb = kk * 4 + kHalf;
      v2f b, a0, a1;
      b.x  = Bp[(size_t)kb * Dk];
      b.y  = Bp[(size_t)(kb + 1) * Dk];
      a0.x = Ht[kb * LSTR + col];
      a0.y = Ht[(kb + 1) * LSTR + col];
      a1.x = Ht[HALF_SZ + kb * LSTR + col];
      a1.y = Ht[HALF_SZ + (kb + 1) * LSTR + col];
      acc0 = __builtin_amdgcn_wmma_f32_16x16x4_f32(
          false, a0, false, b, (short)0, acc0, false, false);
      acc1 = __builtin_amdgcn_wmma_f32_16x16x4_f32(
          false, a1, false, b, (short)0, acc1, false, false);
    }
    const float bias = b2[n0 + col];
#pragma unroll
    for (int v = 0; v < 8; ++v) {
      const int r0 = rowBase + v;
      const int r1 = r0 + 16;
      if (r0 < rows) out[obase[r0] + off + n0 + col] = acc0[v] + bias;
      if (r1 < rows) out[obase[r1] + off + n0 + col] = acc1[v] + bias;
    }
  }
}

// ---------------------------------------------------------------------------
extern "C" void kernel_launch(void* const* d_in, const int* in_sizes, int n_in,
                              void* d_out, int out_size, void* d_ws, size_t ws_size,
                              hipStream_t stream) {
  const float* x  = (const float*)d_in[0];
  const int*   rx = (const int*)d_in[1];
  const int*   rt = (const int*)d_in[2];
  const float* W1 = (const float*)d_in[3];
  const float* b1 = (const float*)d_in[4];

  // setup_inputs() dict order interleaves W2_k/b2_k; detect from in_sizes
  // (interleaved -> in_sizes[6] is b2_0 with 2048 elems).
  const float *W2p[4], *b2p[4];
  const bool interleaved = (n_in > 6 && in_sizes[6] == 2048);
  for (int kk = 0; kk < 4; ++kk) {
    if (interleaved) {
      W2p[kk] = (const float*)d_in[5 + 2 * kk];
      b2p[kk] = (const float*)d_in[6 + 2 * kk];
    } else {
      W2p[kk] = (const float*)d_in[5 + kk];
      b2p[kk] = (const float*)d_in[9 + kk];
    }
  }
  float* out = (float*)d_out;

  int* count = (int*)d_ws;       // 4 ints (padded to 8)
  int* minT  = count + 8;        // 32 ints (padded to 64)
  int* list  = minT + 64;        // 4 * 16384 ints

  init_kernel<<<1, 32, 0, stream>>>(count, minT);
  classify_kernel<<<N_TOK / 256, 256, 0, stream>>>(rx, rt, count, minT, list);

  dim3 grid(N_TOK / M_TILE, K_HEADS);
  fused_mlp_kernel<<<grid, 256, 0, stream>>>(
      x, rt, W1, b1,
      W2p[0], b2p[0], W2p[1], b2p[1], W2p[2], b2p[2], W2p[3], b2p[3],
      count, minT, list, out);
}